// TIMeModel_38500086841944
// MI455X (gfx1250) — compile-verified
//
#include <hip/hip_runtime.h>
#include <hip/hip_bf16.h>

// ---- problem constants (match reference) ----
constexpr int kB  = 128;
constexpr int kS  = 256;
constexpr int kD  = 128;
constexpr int kH  = 4;
constexpr int kDh = 32;
constexpr int kL  = 2;
constexpr int kBS = kB * kS;        // 32768 tokens
constexpr int kNI = 127;            // internal tree nodes (perfect tree of 255)
constexpr float kScale = 0.17677669529663688f; // 1/sqrt(32)
constexpr float kNeg   = -1.0e9f;
constexpr float kEps   = 1e-6f;

typedef float v2f __attribute__((ext_vector_type(2)));
typedef float v8f __attribute__((ext_vector_type(8)));

// GCC-style vector typedefs matching the async builtin signatures exactly
typedef int vi4 __attribute__((__vector_size__(4 * sizeof(int))));
typedef int vi2 __attribute__((__vector_size__(2 * sizeof(int))));

#define AS1 __attribute__((address_space(1)))
#define AS3 __attribute__((address_space(3)))

#if defined(__has_builtin)
# if __has_builtin(__builtin_amdgcn_global_load_async_to_lds_b128)
#  define HAVE_ASYNC_B128 1
# endif
# if __has_builtin(__builtin_amdgcn_global_load_async_to_lds_b64)
#  define HAVE_ASYNC_B64 1
# endif
# if __has_builtin(__builtin_amdgcn_s_wait_asynccnt)
#  define HAVE_WAIT_ASYNC 1
# endif
#endif

// async global -> LDS, 16 bytes per lane (ASYNCcnt-tracked on gfx1250)
__device__ __forceinline__ void ld_async_b128(const float* g, float* l) {
#ifdef HAVE_ASYNC_B128
  __builtin_amdgcn_global_load_async_to_lds_b128((AS1 vi4*)g, (AS3 vi4*)l, 0, 0);
#else
  *(float4*)l = *(const float4*)g;
#endif
}

__device__ __forceinline__ void ld_async_b64(const float* g, float* l) {
#ifdef HAVE_ASYNC_B64
  __builtin_amdgcn_global_load_async_to_lds_b64((AS1 vi2*)g, (AS3 vi2*)l, 0, 0);
#else
  *(float2*)l = *(const float2*)g;
#endif
}

__device__ __forceinline__ void wait_async0() {
#ifdef HAVE_WAIT_ASYNC
  __builtin_amdgcn_s_wait_asynccnt(0);
#else
  asm volatile("s_wait_asynccnt 0x0" ::: "memory");
#endif
}

// fp32 WMMA: D(16x16) = A(16x4) * B(4x16) + C   (exact fp32 path, CDNA5)
__device__ __forceinline__ v8f wmma_f32(v2f a, v2f b, v8f c) {
  return __builtin_amdgcn_wmma_f32_16x16x4_f32(
      /*neg_a=*/false, a, /*neg_b=*/false, b,
      /*c_mod=*/(short)0, c, /*reuse_a=*/false, /*reuse_b=*/false);
}

#define ZERO8 {0.f,0.f,0.f,0.f,0.f,0.f,0.f,0.f}

// ============================================================
// K1: fused Q/K/V projection.  One block = 64 token rows (4 row-tiles),
// 8 waves, wave w owns output columns [16w,16w+16) of Q,K,V.
// B-fragments amortized over 4 row-tiles -> 4x less weight re-read.
// ============================================================
__global__ __launch_bounds__(256) void qkv_kernel(
    const float* __restrict__ x,
    const float* __restrict__ Wq, const float* __restrict__ bq,
    const float* __restrict__ Wk, const float* __restrict__ bk,
    const float* __restrict__ Wv, const float* __restrict__ bv,
    float* __restrict__ Q, float* __restrict__ K, float* __restrict__ V)
{
  __shared__ float lx[64 * kD];               // 32 KB staged activations
  const int tid = threadIdx.x;
  const int r0  = blockIdx.x * 64;

  for (int p = 0; p < 8; ++p) {               // 8192 floats via async B128
    const int flat = (p * 256 + tid) * 4;
    ld_async_b128(x + (size_t)r0 * kD + flat, &lx[flat]);
  }
  wait_async0();
  __syncthreads();

  const int w = tid >> 5, lane = tid & 31;
  const int n0 = w * 16;
  const int am = lane & 15, kh = lane >> 4;   // A-frag: row, k-half
  const int nn = am, mh = kh;                 // B/D-frag: col, row-half

  v8f z = ZERO8;
  v8f aq[4] = {z, z, z, z}, ak[4] = {z, z, z, z}, av[4] = {z, z, z, z};
  for (int k0 = 0; k0 < kD; k0 += 4) {
    const int kr = k0 + 2 * kh;
    v2f bfq, bfk, bfv;
    bfq.x = Wq[(size_t)kr * kD + n0 + nn];
    bfq.y = Wq[(size_t)(kr + 1) * kD + n0 + nn];
    bfk.x = Wk[(size_t)kr * kD + n0 + nn];
    bfk.y = Wk[(size_t)(kr + 1) * kD + n0 + nn];
    bfv.x = Wv[(size_t)kr * kD + n0 + nn];
    bfv.y = Wv[(size_t)(kr + 1) * kD + n0 + nn];
#pragma unroll
    for (int t = 0; t < 4; ++t) {
      v2f a = *(const v2f*)&lx[(t * 16 + am) * kD + k0 + 2 * kh];
      aq[t] = wmma_f32(a, bfq, aq[t]);
      ak[t] = wmma_f32(a, bfk, ak[t]);
      av[t] = wmma_f32(a, bfv, av[t]);
    }
  }

  const float biq = bq[n0 + nn], bik = bk[n0 + nn], biv = bv[n0 + nn];
#pragma unroll
  for (int t = 0; t < 4; ++t)
    for (int r = 0; r < 8; ++r) {
      const size_t row = (size_t)(r0 + t * 16 + r + 8 * mh);
      Q[row * kD + n0 + nn] = aq[t][r] + biq;
      K[row * kD + n0 + nn] = ak[t][r] + bik;
      V[row * kD + n0 + nn] = av[t][r] + biv;
    }
}

// ============================================================
// K2: attention, fused softmax. Block = (16 query rows, head h, batch b).
// Q tile + V head-slice staged via async->LDS. Logits via WMMA (K^T frag
// is contiguous f32x2). Row softmax in LDS. ctx = P*V with K-dim split
// over 4 wave-groups + LDS reduce.
// ============================================================
__global__ __launch_bounds__(256) void attn_kernel(
    const float* __restrict__ Q, const float* __restrict__ Kb,
    const float* __restrict__ V, const float* __restrict__ mask,
    float* __restrict__ CTX, float* __restrict__ wout,
    float* __restrict__ loss, int last)
{
  __shared__ float lq[16 * kDh];    // 2 KB  query tile
  __shared__ float lv[kS * kDh];    // 32 KB V head-slice
  __shared__ float lp[16 * kS];     // 16 KB probability tile
  __shared__ float part[8 * 256];   // 8 KB ctx partials
  __shared__ float red[16 * 16];
  __shared__ float rowstat[16];

  const int q0 = blockIdx.x * 16;
  const int h  = blockIdx.y;
  const int b  = blockIdx.z;
  const int tid = threadIdx.x, w = tid >> 5, lane = tid & 31;
  const int am = lane & 15, kh = lane >> 4;
  const int nn = am, mh = kh;

  // ---- async staging: Q tile (512 floats) + V slice (8192 floats) ----
  {
    const int flat = tid * 2;
    const int row = flat >> 5, col = flat & 31;
    ld_async_b64(Q + (size_t)(b * kS + q0 + row) * kD + h * kDh + col, &lq[flat]);
  }
  for (int p = 0; p < 8; ++p) {
    const int flat = (p * 256 + tid) * 4;
    const int row = flat >> 5, col = flat & 31;
    ld_async_b128(V + (size_t)(b * kS + row) * kD + h * kDh + col, &lv[flat]);
  }
  wait_async0();
  __syncthreads();

  // ---- logits: each wave covers key-tiles w and w+8 ----
  for (int kt = w; kt < 16; kt += 8) {
    v8f acc = ZERO8;
    const float* kbase = Kb + ((size_t)(b * kS + kt * 16)) * kD + h * kDh;
    for (int kd = 0; kd < kDh; kd += 4) {
      v2f a  = *(const v2f*)&lq[am * kDh + kd + 2 * kh];
      v2f bf = *(const v2f*)&kbase[nn * kD + kd + 2 * kh];   // K^T fragment
      acc = wmma_f32(a, bf, acc);
    }
    const int col = kt * 16 + nn;
    const float mv = mask[(size_t)b * kS + col] * kNeg;
    for (int r = 0; r < 8; ++r)
      lp[(r + 8 * mh) * kS + col] = acc[r] * kScale + mv;
  }
  __syncthreads();

  // ---- softmax over 256 keys, 16 threads per row ----
  const int r = tid >> 4, sub = tid & 15;
  float m = -3.0e38f;
  for (int c = sub; c < kS; c += 16) m = fmaxf(m, lp[r * kS + c]);
  red[r * 16 + sub] = m;
  __syncthreads();
  if (sub == 0) {
    float mm = red[r * 16];
    for (int j = 1; j < 16; ++j) mm = fmaxf(mm, red[r * 16 + j]);
    rowstat[r] = mm;
  }
  __syncthreads();
  const float rm = rowstat[r];
  float s = 0.f;
  for (int c = sub; c < kS; c += 16) {
    float e = __expf(lp[r * kS + c] - rm);
    lp[r * kS + c] = e;
    s += e;
  }
  red[r * 16 + sub] = s;
  __syncthreads();
  if (sub == 0) {
    float ss = 0.f;
    for (int j = 0; j < 16; ++j) ss += red[r * 16 + j];
    rowstat[r] = ss;
  }
  __syncthreads();
  const float inv = 1.0f / rowstat[r];
  for (int c = sub; c < kS; c += 16) lp[r * kS + c] *= inv;

  // row S-1: emit attention weights (last layer) + L1 loss (both layers)
  if (q0 == kS - 16 && r == 15) {
    float sn = 0.f;
    for (int c = sub; c < kS; c += 16) {
      const float p = lp[r * kS + c];
      sn += fabsf(p);
      if (last) wout[((size_t)b * kH + h) * kS + c] = p;
    }
    atomicAdd(loss, sn);
  }
  __syncthreads();

  // ---- ctx = P * V : wave w -> out col-tile (w&1), K-slice (w>>1) ----
  const int nt = w & 1, ks = w >> 1;
  v8f acc = ZERO8;
  for (int kd = ks * 64; kd < ks * 64 + 64; kd += 4) {
    v2f a = *(const v2f*)&lp[am * kS + kd + 2 * kh];
    const int kr = kd + 2 * kh;
    v2f bf;
    bf.x = lv[kr * kDh + nt * 16 + nn];
    bf.y = lv[(kr + 1) * kDh + nt * 16 + nn];
    acc = wmma_f32(a, bf, acc);
  }
  for (int rr = 0; rr < 8; ++rr)
    part[w * 256 + (rr + 8 * mh) * 16 + nn] = acc[rr];
  __syncthreads();

  for (int e = tid; e < 512; e += 256) {
    const int t = e >> 8, idx = e & 255;
    const float ssum = part[t * 256 + idx] + part[(t + 2) * 256 + idx] +
                       part[(t + 4) * 256 + idx] + part[(t + 6) * 256 + idx];
    const int mrow = idx >> 4, ncol = idx & 15;
    CTX[((size_t)(b * kS + q0 + mrow)) * kD + h * kDh + t * 16 + ncol] = ssum;
  }
}

// ============================================================
// K3: out-projection + residual + layernorm(g1,b1). 64-row slabs.
// ctx tile staged async; LDS buffer reused for the residual tile.
// ============================================================
__global__ __launch_bounds__(256) void oproj_ln_kernel(
    const float* __restrict__ CTX, const float* __restrict__ Wo,
    const float* __restrict__ bo, const float* __restrict__ xin,
    const float* __restrict__ g1, const float* __restrict__ b1,
    float* __restrict__ out1)
{
  __shared__ float lc[64 * kD];     // 32 KB (ctx tile, then residual tile)
  __shared__ float red[64 * 4];
  __shared__ float red2[64 * 4];
  __shared__ float stat[128];

  const int tid = threadIdx.x;
  const int r0  = blockIdx.x * 64;
  for (int p = 0; p < 8; ++p) {
    const int flat = (p * 256 + tid) * 4;
    ld_async_b128(CTX + (size_t)r0 * kD + flat, &lc[flat]);
  }
  wait_async0();
  __syncthreads();

  const int w = tid >> 5, lane = tid & 31, n0 = w * 16;
  const int am = lane & 15, kh = lane >> 4;
  const int nn = am, mh = kh;

  v8f z = ZERO8;
  v8f acc[4] = {z, z, z, z};
  for (int k0 = 0; k0 < kD; k0 += 4) {
    const int kr = k0 + 2 * kh;
    v2f bf;
    bf.x = Wo[(size_t)kr * kD + n0 + nn];
    bf.y = Wo[(size_t)(kr + 1) * kD + n0 + nn];
#pragma unroll
    for (int t = 0; t < 4; ++t) {
      v2f a = *(const v2f*)&lc[(t * 16 + am) * kD + k0 + 2 * kh];
      acc[t] = wmma_f32(a, bf, acc[t]);
    }
  }
  __syncthreads();                  // all reads of ctx tile complete

  const float bias = bo[n0 + nn];
#pragma unroll
  for (int t = 0; t < 4; ++t)
    for (int r = 0; r < 8; ++r) {
      const int row = t * 16 + r + 8 * mh;
      lc[row * kD + n0 + nn] =
          acc[t][r] + bias + xin[(size_t)(r0 + row) * kD + n0 + nn];
    }
  __syncthreads();

  // layernorm: 4 threads per row over 64 rows
  const int r = tid >> 2, sub = tid & 3;
  float s = 0.f, s2 = 0.f;
  for (int c = sub; c < kD; c += 4) {
    const float v = lc[r * kD + c];
    s += v; s2 += v * v;
  }
  red[r * 4 + sub] = s;
  red2[r * 4 + sub] = s2;
  __syncthreads();
  if (sub == 0) {
    const float a  = red[r * 4] + red[r * 4 + 1] + red[r * 4 + 2] + red[r * 4 + 3];
    const float bb = red2[r * 4] + red2[r * 4 + 1] + red2[r * 4 + 2] + red2[r * 4 + 3];
    const float mu  = a * (1.0f / kD);
    const float var = bb * (1.0f / kD) - mu * mu;
    stat[r]      = mu;
    stat[64 + r] = rsqrtf(var + kEps);
  }
  __syncthreads();
  const float mu = stat[r], rs = stat[64 + r];
  for (int c = sub; c < kD; c += 4)
    out1[(size_t)(r0 + r) * kD + c] = g1[c] * (lc[r * kD + c] - mu) * rs + b1[c];
}

// ============================================================
// K4: one bottom-up tree level. Updates internal nodes [base, base+count):
//     tab[n] = tanh(child_l + child_r). Leaf children (>=127) read out1.
// ============================================================
__global__ void tree_level_kernel(const float* __restrict__ out1,
                                  float* __restrict__ tab,
                                  int base, int count)
{
  const int i = blockIdx.x * blockDim.x + threadIdx.x;
  const int total = kB * count * kD;
  if (i >= total) return;
  const int c = i & 127;
  const int t = i >> 7;
  const int node = base + (t % count);
  const int b = t / count;
  const int c1 = 2 * node + 1, c2 = 2 * node + 2;
  const float v1 = (c1 < kNI) ? tab[((size_t)b * kNI + c1) * kD + c]
                              : out1[((size_t)b * kS + c1) * kD + c];
  const float v2 = (c2 < kNI) ? tab[((size_t)b * kNI + c2) * kD + c]
                              : out1[((size_t)b * kS + c2) * kD + c];
  tab[((size_t)b * kNI + node) * kD + c] = tanhf(v1 + v2);
}

// ============================================================
// K5: x_next = LN(out1 + tree_agg, g2, b2). Leaves & pass-through row
//     have tab == out1 (unchanged), so use 2*out1 there.
// ============================================================
__global__ __launch_bounds__(256) void ln2_kernel(
    const float* __restrict__ out1, const float* __restrict__ tab,
    const float* __restrict__ g2, const float* __restrict__ b2,
    float* __restrict__ xout)
{
  __shared__ float red[16 * 16];
  __shared__ float red2[16 * 16];
  __shared__ float stat[32];

  const int tid = threadIdx.x;
  const int r = tid >> 4, sub = tid & 15;
  const int row = blockIdx.x * 16 + r;
  const int b = row >> 8, s_ = row & 255;

  float vals[8];
  float s = 0.f, s2 = 0.f;
  for (int j = 0; j < 8; ++j) {
    const int c = sub + 16 * j;
    const float o = out1[(size_t)row * kD + c];
    const float t = (s_ < kNI) ? tab[((size_t)b * kNI + s_) * kD + c] : o;
    const float v = o + t;
    vals[j] = v;
    s += v; s2 += v * v;
  }
  red[r * 16 + sub] = s;
  red2[r * 16 + sub] = s2;
  __syncthreads();
  if (sub == 0) {
    float a = 0.f, bb = 0.f;
    for (int j = 0; j < 16; ++j) { a += red[r * 16 + j]; bb += red2[r * 16 + j]; }
    const float mu  = a * (1.0f / kD);
    const float var = bb * (1.0f / kD) - mu * mu;
    stat[r]      = mu;
    stat[16 + r] = rsqrtf(var + kEps);
  }
  __syncthreads();
  const float mu = stat[r], rs = stat[16 + r];
  for (int j = 0; j < 8; ++j) {
    const int c = sub + 16 * j;
    xout[(size_t)row * kD + c] = g2[c] * (vals[j] - mu) * rs + b2[c];
  }
}

// ============================================================
extern "C" void kernel_launch(void* const* d_in, const int* in_sizes, int n_in,
                              void* d_out, int out_size, void* d_ws, size_t ws_size,
                              hipStream_t stream) {
  (void)in_sizes; (void)n_in; (void)out_size; (void)ws_size;

  const float* x0   = (const float*)d_in[0];
  const float* mask = (const float*)d_in[1];
  const float* Wq = (const float*)d_in[5];
  const float* bq = (const float*)d_in[6];
  const float* Wk = (const float*)d_in[7];
  const float* bk = (const float*)d_in[8];
  const float* Wv = (const float*)d_in[9];
  const float* bv = (const float*)d_in[10];
  const float* Wo = (const float*)d_in[11];
  const float* bo = (const float*)d_in[12];
  const float* g1 = (const float*)d_in[13];
  const float* b1 = (const float*)d_in[14];
  const float* g2 = (const float*)d_in[15];
  const float* b2 = (const float*)d_in[16];

  float* out_x = (float*)d_out;
  float* loss  = out_x + (size_t)kBS * kD;   // total_l1 (scalar)
  float* wout  = loss + 1;                   // weights (B,H,S)

  float* ws = (float*)d_ws;
  const size_t SZ = (size_t)kBS * kD;        // 4.19 M floats per activation buf
  float* Qb = ws;
  float* Kb = ws + SZ;
  float* Vb = ws + 2 * SZ;
  float* Cb = ws + 3 * SZ;
  float* O1 = ws + 4 * SZ;
  float* XB = ws + 5 * SZ;
  float* TB = ws + 6 * SZ;                   // internal tree nodes: B*127*D

  (void)hipMemsetAsync(loss, 0, sizeof(float), stream);

  const float* xin = x0;
  for (int l = 0; l < kL; ++l) {
    const float* wq = Wq + (size_t)l * kD * kD;
    const float* wk = Wk + (size_t)l * kD * kD;
    const float* wv = Wv + (size_t)l * kD * kD;
    const float* wo = Wo + (size_t)l * kD * kD;
    const float* bql = bq + (size_t)l * kD;
    const float* bkl = bk + (size_t)l * kD;
    const float* bvl = bv + (size_t)l * kD;
    const float* bol = bo + (size_t)l * kD;
    const float* g1l = g1 + (size_t)l * kD;
    const float* b1l = b1 + (size_t)l * kD;
    const float* g2l = g2 + (size_t)l * kD;
    const float* b2l = b2 + (size_t)l * kD;
    const int last = (l == kL - 1);
    float* xout = last ? out_x : XB;

    qkv_kernel<<<kBS / 64, 256, 0, stream>>>(xin, wq, bql, wk, bkl, wv, bvl,
                                             Qb, Kb, Vb);
    attn_kernel<<<dim3(kS / 16, kH, kB), 256, 0, stream>>>(
        Qb, Kb, Vb, mask, Cb, wout, loss, last);
    oproj_ln_kernel<<<kBS / 64, 256, 0, stream>>>(Cb, wo, bol, xin, g1l, b1l, O1);

    for (int d = 7; d >= 1; --d) {           // bottom-up tree levels
      const int dd = d - 1;
      const int base = (1 << dd) - 1;
      const int count = 1 << dd;
      const int total = kB * count * kD;
      tree_level_kernel<<<(total + 255) / 256, 256, 0, stream>>>(O1, TB, base,
                                                                 count);
    }
    ln2_kernel<<<kBS / 16, 256, 0, stream>>>(O1, TB, g2l, b2l, xout);
    xin = xout;
  }
}